// DGPLoss_4105988735223
// MI455X (gfx1250) — compile-verified
//
#include <hip/hip_runtime.h>
#include <hip/hip_bf16.h>
#include <cstdint>

typedef float v2f __attribute__((ext_vector_type(2)));
typedef float v8f __attribute__((ext_vector_type(8)));

#define EPSF 1e-8f
#define TAUF 10.0f

#define B_  4
#define C_  64
#define H_  512
#define W_  512
#define PH_ 5
#define HV  510              // valid region (102 patches * 5)
#define NWG 32               // 16-pixel groups per row (covers 0..511, mask w<510)
#define WPB 8                // waves per block

// One wave handles 16 consecutive pixels of one row of one image.
// seg_sq for the 16 pixels is computed as the diagonal of Gram(diff) via
// 16 chained V_WMMA_F32_16X16X4_F32 ops over K = 64 channels.
__global__ __launch_bounds__(256) void dgp_main(const float* __restrict__ seg,
                                                const float* __restrict__ dep,
                                                float* __restrict__ ws) {
    const int  tid   = threadIdx.x;
    const int  lane  = tid & 31;
    const long waveId = (long)blockIdx.x * WPB + (tid >> 5);

    const int wg = (int)(waveId & (NWG - 1));        // w-group 0..31
    const int h  = (int)((waveId >> 5) % HV);        // row 0..509
    const int b  = (int)(waveId / ((long)NWG * HV)); // image 0..3

    const int pix  = lane & 15;                      // pixel index in group (M and N)
    const int half = lane >> 4;                      // lane-half selects K pair {0,1} vs {2,3}
    const int w    = wg * 16 + pix;                  // 0..511 (always in-bounds of W=512)
    const int wl   = (w < HV - 1) ? w : (HV - 1);    // clamp only for center-index math
    const int hc   = (h  / PH_) * PH_ + 2;           // patch-center row
    const int wc   = (wl / PH_) * PH_ + 2;           // patch-center col

    const long chanStride = (long)H_ * W_;
    const float* segb = seg + (long)b * C_ * chanStride;
    const long pixOff = (long)h  * W_ + w;
    const long cenOff = (long)hc * W_ + wc;

    v8f acc = {0.f, 0.f, 0.f, 0.f, 0.f, 0.f, 0.f, 0.f};

#pragma unroll
    for (int kb = 0; kb < C_; kb += 4) {
        // 32-bit A 16x4 layout: lanes 0-15 -> V0:K=kb,   V1:K=kb+1
        //                       lanes16-31 -> V0:K=kb+2, V1:K=kb+3
        // B 4x16 layout matches the same register image, so d feeds both operands.
        const int ch0 = kb + 2 * half;
        const int ch1 = ch0 + 1;
        const float p0 = segb[(long)ch0 * chanStride + pixOff];
        const float p1 = segb[(long)ch1 * chanStride + pixOff];
        const float c0 = segb[(long)ch0 * chanStride + cenOff];
        const float c1 = segb[(long)ch1 * chanStride + cenOff];
        v2f d;
        d.x = c0 - p0;
        d.y = c1 - p1;
        acc = __builtin_amdgcn_wmma_f32_16x16x4_f32(
                  /*neg_a=*/false, d, /*neg_b=*/false, d,
                  /*c_mod=*/(short)0, acc, /*reuse_a=*/false, /*reuse_b=*/false);
    }

    // Diagonal ownership (C/D layout: VGPR j holds M=j (lanes 0-15) / M=8+j (lanes 16-31)):
    //   lanes 0-7  own pixel = lane      -> acc[lane]
    //   lanes 24-31 own pixel = lane-16  -> acc[lane-24]
    const int  idx   = lane & 7;
    const bool owner = (lane < 8) || (lane >= 24);
    float segsq = acc[0];
    segsq = (idx == 1) ? acc[1] : segsq;
    segsq = (idx == 2) ? acc[2] : segsq;
    segsq = (idx == 3) ? acc[3] : segsq;
    segsq = (idx == 4) ? acc[4] : segsq;
    segsq = (idx == 5) ? acc[5] : segsq;
    segsq = (idx == 6) ? acc[6] : segsq;
    segsq = (idx == 7) ? acc[7] : segsq;

    // Depth branch for this lane's pixel (h, w).
    const float* depb = dep + (long)b * chanStride;
    const float dpix  = depb[pixOff];
    const float dcen  = depb[cenOff];
    const float ddiff = fabsf(dcen - dpix);

    const bool isCenter = (h == hc) && (w == wc);
    const bool mask = owner && (w < HV)
                   && (ddiff > EPSF)
                   && (segsq > EPSF * EPSF)      // sqrt(seg_sq) > EPS
                   && (dpix  > EPSF)
                   && !isCenter;

    const float loss = expf(-ddiff / TAUF) * expf(-segsq);
    const float lsum = mask ? loss : 0.0f;
    const float lcnt = mask ? 1.0f : 0.0f;

    // Block reduction in LDS, then one atomic pair per block.
    __shared__ float s_sum[256];
    __shared__ float s_cnt[256];
    s_sum[tid] = lsum;
    s_cnt[tid] = lcnt;
    __syncthreads();
#pragma unroll
    for (int s = 128; s > 0; s >>= 1) {
        if (tid < s) {
            s_sum[tid] += s_sum[tid + s];
            s_cnt[tid] += s_cnt[tid + s];
        }
        __syncthreads();
    }
    if (tid == 0) {
        atomicAdd(&ws[0], s_sum[0]);
        atomicAdd(&ws[1], s_cnt[0]);
    }
}

__global__ void dgp_finalize(const float* __restrict__ ws, float* __restrict__ out) {
    const float n = (ws[1] < 1.0f) ? 1.0f : ws[1];
    out[0] = ws[0] / n;
}

extern "C" void kernel_launch(void* const* d_in, const int* in_sizes, int n_in,
                              void* d_out, int out_size, void* d_ws, size_t ws_size,
                              hipStream_t stream) {
    (void)in_sizes; (void)n_in; (void)out_size; (void)ws_size;
    const float* seg = (const float*)d_in[0];   // [4, 64, 512, 512] f32
    const float* dep = (const float*)d_in[1];   // [4, 1, 512, 512] f32
    float* out = (float*)d_out;                 // scalar f32
    float* ws  = (float*)d_ws;                  // [sum, count]

    hipMemsetAsync(d_ws, 0, 2 * sizeof(float), stream);

    const long totalWaves = (long)B_ * HV * NWG;        // 4*510*32 = 65280
    const int  blocks     = (int)(totalWaves / WPB);    // 8160
    dgp_main<<<blocks, 256, 0, stream>>>(seg, dep, ws);
    dgp_finalize<<<1, 1, 0, stream>>>(ws, out);
}